// SGC_4647154614446
// MI455X (gfx1250) — compile-verified
//
#include <hip/hip_runtime.h>
#include <hip/hip_bf16.h>

typedef __attribute__((ext_vector_type(2))) float v2f;
typedef __attribute__((ext_vector_type(8))) float v8f;

#define N_FEAT 128
#define N_CLS  64

// Hardware fp32 atomic add (GLOBAL_ATOMIC_ADD_F32), avoiding the default CAS loop.
__device__ __forceinline__ void atomic_fadd_hw(float* addr, float val) {
    unsafeAtomicAdd(addr, val);
}

// ---------------- zero scratch ----------------
__global__ void sgc_zero(float* __restrict__ p, long long n) {
    long long i = (long long)blockIdx.x * blockDim.x + threadIdx.x;
    if (i < n) p[i] = 0.0f;
}

// ---------------- degree accumulation (in-edges) ----------------
__global__ void sgc_deg(const int* __restrict__ dst, float* __restrict__ deg, int E) {
    int i = blockIdx.x * blockDim.x + threadIdx.x;
    if (i < E) atomic_fadd_hw(&deg[dst[i]], 1.0f);   // integer-valued: exact
}

// ---------------- deg -> deg^{-1/2} (self-loop adds 1) ----------------
__global__ void sgc_dinv(float* __restrict__ deg, int N) {
    int i = blockIdx.x * blockDim.x + threadIdx.x;
    if (i < N) deg[i] = rsqrtf(deg[i] + 1.0f);   // deg+1 >= 1 always
}

// ---------------- SpMM hop: y[dst] += dinv[src]*dinv[dst]*x[src] ----------------
// 32 lanes per edge, float4 (=4 feats) per lane -> 128 feats.
// Edges e in [0,E) are real edges; e in [E,E+N) are self-loops.
__global__ void sgc_spmm(const int* __restrict__ src, const int* __restrict__ dst,
                         const float* __restrict__ dinv,
                         const float* __restrict__ xin, float* __restrict__ xout,
                         int E, int N) {
    long long gid = (long long)blockIdx.x * blockDim.x + threadIdx.x;
    long long e   = gid >> 5;
    int lane      = threadIdx.x & 31;
    if (e >= (long long)E + N) return;
    int s, d;
    if (e < E) { s = src[e]; d = dst[e]; }
    else       { s = d = (int)(e - E); }
    float nrm = dinv[s] * dinv[d];
    const float4* xi = (const float4*)(xin + (long long)s * N_FEAT);
    float*        yo = xout + (long long)d * N_FEAT;
    float4 v = xi[lane];
    int f = lane * 4;
    atomic_fadd_hw(&yo[f + 0], nrm * v.x);
    atomic_fadd_hw(&yo[f + 1], nrm * v.y);
    atomic_fadd_hw(&yo[f + 2], nrm * v.z);
    atomic_fadd_hw(&yo[f + 3], nrm * v.w);
}

// ---------------- GEMM via V_WMMA_F32_16X16X4_F32 ----------------
// logits[m][n] = sum_k H[m][k] * W[n][k] + b[n]
// One wave computes one 16x16 (M x N) tile; 4 waves/block cover N=64.
// fp32 A 16x4 layout: lanes 0-15 hold M=lane, {v0=K0, v1=K1}; lanes 16-31 hold {K2,K3}.
// fp32 B 4x16 layout (mirror): lanes 0-15 hold N=lane, {v0=K0, v1=K1}; lanes 16-31 {K2,K3}.
// fp32 C/D: vgpr r -> M = r + 8*(lane>=16), N = lane&15.
__global__ __launch_bounds__(128) void sgc_gemm_wmma(
        const float* __restrict__ H, const float* __restrict__ W,
        const float* __restrict__ bias, float* __restrict__ out) {
    const int lane = threadIdx.x & 31;
    const int wave = threadIdx.x >> 5;
    const int m0   = blockIdx.x * 16;
    const int n0   = wave * 16;
    const int half = lane >> 4;      // 0: lanes 0-15, 1: lanes 16-31
    const int l    = lane & 15;

    const float* arow = H + (long long)(m0 + l) * N_FEAT;  // A row M = l
    const float* brow = W + (long long)(n0 + l) * N_FEAT;  // B col N = l  (W row-major [cls][feat])

    v8f c = {};
#pragma unroll
    for (int k = 0; k < N_FEAT; k += 4) {
        const int ka = k + half * 2;                      // even -> 8B aligned
        v2f a = *(const v2f*)(arow + ka);
        v2f b = *(const v2f*)(brow + ka);
        c = __builtin_amdgcn_wmma_f32_16x16x4_f32(
                /*neg_a=*/false, a, /*neg_b=*/false, b,
                /*c_mod=*/(short)0, c, /*reuse_a=*/false, /*reuse_b=*/false);
    }

    const int col = n0 + l;
    const float bv = bias[col];
#pragma unroll
    for (int r = 0; r < 8; ++r) {
        const int row = m0 + r + half * 8;
        out[(long long)row * N_CLS + col] = c[r] + bv;
    }
}

// ---------------- row-wise log_softmax, in place, one wave per node ----------------
__global__ void sgc_logsoftmax(float* __restrict__ out, int N) {
    int node = blockIdx.x * (blockDim.x >> 5) + (threadIdx.x >> 5);
    if (node >= N) return;
    int lane = threadIdx.x & 31;
    float* row = out + (long long)node * N_CLS;
    float v0 = row[lane];
    float v1 = row[lane + 32];
    float m = fmaxf(v0, v1);
#pragma unroll
    for (int off = 16; off > 0; off >>= 1)
        m = fmaxf(m, __shfl_xor(m, off, 32));
    float s = expf(v0 - m) + expf(v1 - m);
#pragma unroll
    for (int off = 16; off > 0; off >>= 1)
        s += __shfl_xor(s, off, 32);
    float lse = m + logf(s);
    row[lane]      = v0 - lse;
    row[lane + 32] = v1 - lse;
}

extern "C" void kernel_launch(void* const* d_in, const int* in_sizes, int n_in,
                              void* d_out, int out_size, void* d_ws, size_t ws_size,
                              hipStream_t stream) {
    const float* x  = (const float*)d_in[0];
    const int*   ei = (const int*)d_in[1];
    const float* W  = (const float*)d_in[2];
    const float* b  = (const float*)d_in[3];

    const int N = in_sizes[0] / N_FEAT;   // 50000
    const int E = in_sizes[1] / 2;        // 800000
    const int* src = ei;
    const int* dst = ei + E;

    float* ws   = (float*)d_ws;
    float* dinv = ws;                                  // N floats (deg then dinv)
    float* h1   = ws + N;                              // N*128
    float* h2   = h1 + (long long)N * N_FEAT;          // N*128
    float* out  = (float*)d_out;

    const long long nzero = (long long)N * (1 + 2 * N_FEAT); // contiguous region
    {
        long long blocks = (nzero + 255) / 256;
        sgc_zero<<<(unsigned)blocks, 256, 0, stream>>>(ws, nzero);
    }
    sgc_deg<<<(E + 255) / 256, 256, 0, stream>>>(dst, dinv, E);
    sgc_dinv<<<(N + 255) / 256, 256, 0, stream>>>(dinv, N);

    {
        long long threads = ((long long)E + N) * 32;
        long long blocks  = (threads + 255) / 256;
        sgc_spmm<<<(unsigned)blocks, 256, 0, stream>>>(src, dst, dinv, x,  h1, E, N);
        sgc_spmm<<<(unsigned)blocks, 256, 0, stream>>>(src, dst, dinv, h1, h2, E, N);
    }

    sgc_gemm_wmma<<<N / 16, 128, 0, stream>>>(h2, W, b, out);

    sgc_logsoftmax<<<(N + 7) / 8, 256, 0, stream>>>(out, N);
}